// RCVHCWriter_30889404793251
// MI455X (gfx1250) — compile-verified
//
#include <hip/hip_runtime.h>

typedef __attribute__((ext_vector_type(2))) float v2f;
typedef __attribute__((ext_vector_type(8))) float v8f;

#define H     2048
#define SEQ   4096
#define MDIM  256
#define BLK   64
#define NB    64
#define EPSF  1e-6f
#define KB    128

// ---------------- K1: per-seq-block means, concat layout blk[64][4096] ----
// blk[b][0:2048]   = mean over 64 rows of h_in  block b
// blk[b][2048:4096]= mean over 64 rows of h_out block b
__global__ void k_blk_mean(const float* __restrict__ h_in,
                           const float* __restrict__ h_out,
                           float* __restrict__ blk) {
  int t = blockIdx.x * blockDim.x + threadIdx.x;   // 64*4096 threads
  int b = t >> 12;
  int j = t & 4095;
  const float* src = (j < H) ? (h_in  + (size_t)b * BLK * H + j)
                             : (h_out + (size_t)b * BLK * H + (j - H));
  float s = 0.f;
  #pragma unroll 8
  for (int r = 0; r < BLK; ++r) s += src[(size_t)r * H];
  blk[t] = s * (1.0f / BLK);
}

// ---------------- K2: attn head-mean, block-column sum, future mask -------
// w[b][s] = (1/8) * sum_h sum_{k<64} attn[h][s][b*64+k]   if s >= (b+1)*64
// One wave per (b,s). Masked region written as 0 (never read from HBM).
// Non-temporal: 512MB stream must not evict L2-resident h_out/weights.
__global__ void k_attn_w(const float* __restrict__ attn, float* __restrict__ w) {
  int wave = threadIdx.x >> 5;
  int lane = threadIdx.x & 31;
  int p = blockIdx.x * 8 + wave;     // 64*4096 pairs total
  int b = p >> 12;
  int s = p & 4095;
  float acc = 0.f;
  if (s >= (b + 1) * BLK) {
    const float* base = attn + (size_t)s * SEQ + b * BLK + lane * 2;
    #pragma unroll
    for (int h = 0; h < 8; ++h) {
      v2f v = __builtin_nontemporal_load(
          reinterpret_cast<const v2f*>(base + (size_t)h * SEQ * SEQ));
      acc += v.x + v.y;
    }
  }
  #pragma unroll
  for (int off = 16; off > 0; off >>= 1)
    acc += __shfl_down(acc, off, 32);
  if (lane == 0) w[(size_t)b * SEQ + s] = acc * 0.125f;
}

// ---------------- K3: usage[b] = sum_s w[b][s]; usage_mass output --------
__global__ void k_usage(const float* __restrict__ w, float* __restrict__ usage,
                        float* __restrict__ usage_mass) {
  __shared__ float red[256];
  int b = blockIdx.x;
  float s = 0.f;
  for (int i = threadIdx.x; i < SEQ; i += 256) s += w[(size_t)b * SEQ + i];
  red[threadIdx.x] = s;
  __syncthreads();
  for (int st = 128; st > 0; st >>= 1) {
    if ((int)threadIdx.x < st) red[threadIdx.x] += red[threadIdx.x + st];
    __syncthreads();
  }
  if (threadIdx.x == 0) {
    float u = red[0];
    usage[b] = u;
    usage_mass[b] = (u > EPSF) ? u : 0.f;
  }
}

// ---------------- K4: generic fp32 WMMA GEMM (compile-time specialized) ---
// out[64 x N] = A[64 x K] @ B (+ bias / usage-normalize)
//   BROW=1 : B[k][n] = B[k*H + n]   (w @ h_out; stride hardcoded to H)
//   BROW=0 : B[k][n] = B[n*ldb + k] (X @ W^T case, ldb = K)
// MODE 0: out = acc + bias[n]
// MODE 1: out = (acc + usage[m]*bias[n]) / (usage[m] + eps)
// MODE 2: out = acc
// Grid: x = 4 (M-tiles of 16), y = N/128 (8 waves x 16 cols). Block = 256.
template <int BROW, int MODE>
__global__ void k_gemm_wmma(const float* __restrict__ A, int lda,
                            const float* __restrict__ B, int ldb,
                            int K, const float* __restrict__ bias,
                            float* __restrict__ out, int ldo,
                            const float* __restrict__ usage) {
  // +2 pad: rows 8B-aligned (b64 LDS reads) and stride 130 -> banks 2*l,
  // conflict-free across lanes 0..15.
  __shared__ float As[16][KB + 2];
  int tid  = threadIdx.x;
  int wave = tid >> 5;
  int lane = tid & 31;
  int half = lane >> 4;
  int l    = lane & 15;
  int mrow0 = blockIdx.x * 16;
  int col   = (blockIdx.y * 8 + wave) * 16 + l;

  v8f c = {0.f, 0.f, 0.f, 0.f, 0.f, 0.f, 0.f, 0.f};

  for (int kb = 0; kb < K; kb += KB) {
    // stage A[16][128] via float4 loads (2 per thread), coalesced
    #pragma unroll
    for (int i = tid; i < 16 * (KB / 4); i += 256) {
      int m  = i >> 5;                 // 32 float4 per row
      int k4 = i & 31;
      float4 val = reinterpret_cast<const float4*>(
          A + (size_t)(mrow0 + m) * lda + kb)[k4];
      float* dst = &As[m][k4 * 4];
      dst[0] = val.x; dst[1] = val.y; dst[2] = val.z; dst[3] = val.w;
    }
    __syncthreads();

    // per-lane operand pointers for this K-block (incremental addressing)
    const float* a_ptr = &As[l][2 * half];
    const float* b_ptr = BROW ? (B + (size_t)(kb + 2 * half) * H + col)
                              : (B + (size_t)col * ldb + kb + 2 * half);
    #pragma unroll 8
    for (int k0 = 0; k0 < KB; k0 += 4) {
      v2f a, bb;
      a.x = a_ptr[0];
      a.y = a_ptr[1];
      if (BROW) {                        // two rows of B, same column
        bb.x = b_ptr[0];
        bb.y = b_ptr[H];
      } else {                           // adjacent k of W row -> b64 load
        bb = *reinterpret_cast<const v2f*>(b_ptr);
      }
      c = __builtin_amdgcn_wmma_f32_16x16x4_f32(false, a, false, bb,
                                                (short)0, c, false, false);
      a_ptr += 4;
      b_ptr += BROW ? 4 * H : 4;
    }
    __syncthreads();
  }

  #pragma unroll
  for (int v = 0; v < 8; ++v) {          // D: VGPR v -> rows v / v+8
    int row = mrow0 + v + 8 * half;
    float acc = c[v];
    if (MODE == 0) {
      acc += bias[col];
    } else if (MODE == 1) {
      float us = usage[row];
      acc = (acc + us * bias[col]) / (us + EPSF);
    }
    out[(size_t)row * ldo + col] = acc;
  }
}

// ---------------- K5: delta = masked nan_to_num(rmsnorm(v2 - c_self)) -----
__global__ void k_delta(const float* __restrict__ v2,
                        const float* __restrict__ c_self,
                        const float* __restrict__ norm_w,
                        const float* __restrict__ usage,
                        float* __restrict__ delta) {
  __shared__ float red[256];
  int b = blockIdx.x, c = threadIdx.x;
  float x = v2[b * MDIM + c] - c_self[b * MDIM + c];
  red[c] = x * x;
  __syncthreads();
  for (int st = 128; st > 0; st >>= 1) {
    if (c < st) red[c] += red[c + st];
    __syncthreads();
  }
  float ms = red[0] * (1.0f / MDIM);
  float y = x * rsqrtf(ms + EPSF) * norm_w[c];
  if (__builtin_isnan(y)) y = 0.f;
  else if (__builtin_isinf(y)) y = (y > 0.f) ? 3.402823466e38f : -3.402823466e38f;
  if (!(usage[b] > EPSF)) y = 0.f;
  delta[b * MDIM + c] = y;
}

extern "C" void kernel_launch(void* const* d_in, const int* in_sizes, int n_in,
                              void* d_out, int out_size, void* d_ws, size_t ws_size,
                              hipStream_t stream) {
  (void)in_sizes; (void)n_in; (void)out_size; (void)ws_size;
  const float* h_in  = (const float*)d_in[0];
  const float* h_out = (const float*)d_in[1];
  const float* attn  = (const float*)d_in[2];
  const float* Wrk = (const float*)d_in[3];  const float* brk = (const float*)d_in[4];
  const float* Wak = (const float*)d_in[5];  const float* bak = (const float*)d_in[6];
  const float* Wrv = (const float*)d_in[7];  const float* brv = (const float*)d_in[8];
  const float* Wsp = (const float*)d_in[9];  const float* bsp = (const float*)d_in[10];
  const float* Wup = (const float*)d_in[11]; const float* bup = (const float*)d_in[12];
  const float* Wdq = (const float*)d_in[13]; const float* bdq = (const float*)d_in[14];
  const float* Wdk = (const float*)d_in[15]; const float* bdk = (const float*)d_in[16];
  const float* Wdv = (const float*)d_in[17]; const float* bdv = (const float*)d_in[18];
  const float* norm_w = (const float*)d_in[19];
  float* out = (float*)d_out;

  // workspace layout (floats): ~2.75 MB total
  float* ws     = (float*)d_ws;
  float* blk    = ws;                  // 64*4096 concat means
  float* w      = blk   + NB * 4096;   // 64*4096 masked attn weights
  float* usage  = w     + NB * 4096;   // 64
  float* u      = usage + NB;          // 64*2048  (= w @ h_out)
  float* c_self = u     + NB * H;      // 64*256
  float* v2     = c_self+ NB * MDIM;   // 64*256
  float* delta  = v2    + NB * MDIM;   // 64*256

  k_blk_mean<<<1024, 256, 0, stream>>>(h_in, h_out, blk);
  k_attn_w<<<(NB * SEQ) / 8, 256, 0, stream>>>(attn, w);
  k_usage<<<NB, 256, 0, stream>>>(w, usage, out + 98304);

  // u = w @ h_out   [64 x 2048], K = 4096, B row-major (stride H)
  k_gemm_wmma<1, 2><<<dim3(4, 16), 256, 0, stream>>>(w, SEQ, h_out, H, SEQ,
                                                     nullptr, u, H, nullptr);
  // projections of h_out_blk (blk + 2048, row stride 4096), K = 2048
  k_gemm_wmma<0, 0><<<dim3(4, 2), 256, 0, stream>>>(blk + H, 4096, Wrk, H, H,
                                                    brk, out + 0,     MDIM, nullptr);
  k_gemm_wmma<0, 0><<<dim3(4, 2), 256, 0, stream>>>(blk + H, 4096, Wak, H, H,
                                                    bak, out + 16384, MDIM, nullptr);
  k_gemm_wmma<0, 0><<<dim3(4, 2), 256, 0, stream>>>(blk + H, 4096, Wrv, H, H,
                                                    brv, out + 32768, MDIM, nullptr);
  // c_self = concat(h_in_blk, h_out_blk) @ Wsp^T, K = 4096
  k_gemm_wmma<0, 0><<<dim3(4, 2), 256, 0, stream>>>(blk, 4096, Wsp, 2 * H, 2 * H,
                                                    bsp, c_self, MDIM, nullptr);
  // v2 = (u @ Wup^T + usage*bup) / (usage + eps), K = 2048
  k_gemm_wmma<0, 1><<<dim3(4, 2), 256, 0, stream>>>(u, H, Wup, H, H,
                                                    bup, v2, MDIM, usage);

  k_delta<<<NB, 256, 0, stream>>>(v2, c_self, norm_w, usage, delta);

  // delta projections, K = 256
  k_gemm_wmma<0, 0><<<dim3(4, 2), 256, 0, stream>>>(delta, MDIM, Wdq, MDIM, MDIM,
                                                    bdq, out + 49152, MDIM, nullptr);
  k_gemm_wmma<0, 0><<<dim3(4, 2), 256, 0, stream>>>(delta, MDIM, Wdk, MDIM, MDIM,
                                                    bdk, out + 65536, MDIM, nullptr);
  k_gemm_wmma<0, 0><<<dim3(4, 2), 256, 0, stream>>>(delta, MDIM, Wdv, MDIM, MDIM,
                                                    bdv, out + 81920, MDIM, nullptr);
}